// LocallyConnectedLinear_19155554140908
// MI455X (gfx1250) — compile-verified
//
#include <hip/hip_runtime.h>

// Locally-connected layer, MI455X (gfx1250, wave32).
// out[b,h,w,o] = sum_k x_patch[b,h,w,k] * W[h,w,k,o],  k = c*9 + kh*3 + kw
// Per (h,w): [8x576] x [576x64] GEMM, fp32, via V_WMMA_F32_16X16X4_F32.
// Memory-bound (~3.9 FLOP/B, 132MB weights streamed once, L2-resident on replay).

typedef __attribute__((ext_vector_type(2))) float v2f;
typedef __attribute__((ext_vector_type(8))) float v8f;

#define H_OUT 30
#define W_OUT 30
#define CIN   64
#define COUT  64
#define KTOT  576            // 64 * 3 * 3
#define ASTR  580            // LDS row stride (floats): 580%64=4 -> conflict-free, 8B aligned
#define BATCH 8

__launch_bounds__(128)
__global__ void lc_wmma_kernel(const float* __restrict__ x,
                               const float* __restrict__ w,
                               float* __restrict__ out)
{
    __shared__ float Alds[16 * ASTR];   // 37,120 B of 320 KB WGP LDS

    const int loc  = blockIdx.x;        // 0..899  -> (h, wcol)
    const int h    = loc / W_OUT;
    const int wcol = loc - h * W_OUT;
    const int tid  = threadIdx.x;       // 0..127 (4 waves)

    // ---- Stage A = im2col patch [16 x 576] into LDS (rows 8..15 zero-padded M) ----
    // x is NHWC (8,32,32,64); k ordered C-slowest: k = c*9 + kh*3 + kw
    for (int idx = tid; idx < 16 * KTOT; idx += 128) {
        const int m = idx / KTOT;
        const int k = idx - m * KTOT;
        float v = 0.0f;
        if (m < BATCH) {
            const int c  = k / 9;
            const int r  = k - c * 9;
            const int kh = r / 3;
            const int kw = r - kh * 3;
            v = x[(((size_t)m * 32 + (h + kh)) * 32 + (wcol + kw)) * CIN + c];
        }
        Alds[m * ASTR + k] = v;
    }
    __syncthreads();

    // ---- Per-wave 16x16 output tile over K=576 ----
    const int lane = tid & 31;
    const int wid  = tid >> 5;          // 0..3 -> N tile
    const int n0   = wid * 16;
    const int koff = (lane >> 4) * 2;   // 0 (lanes 0-15) or 2 (lanes 16-31)
    const int mrow = lane & 15;         // A: M index ; B/D: N index

    // Weight base for this lane's column; row k lives at stride COUT floats.
    const float* __restrict__ Wp = w + (size_t)loc * KTOT * COUT + n0 + mrow;
    const float* __restrict__ Ap = Alds + mrow * ASTR + koff;

    v8f acc = {};
    #pragma unroll 4
    for (int k0 = 0; k0 < KTOT; k0 += 4) {
        // A frag (16x4 f32): VGPR0 = A[m][k0+koff], VGPR1 = A[m][k0+koff+1]
        const v2f a = *(const v2f*)(Ap + k0);

        // B frag (4x16 f32): VGPR0 = W[k0+koff][n], VGPR1 = W[k0+koff+1][n]
        v2f b;
        b.x = Wp[(size_t)(k0 + koff) * COUT];
        b.y = Wp[(size_t)(k0 + koff + 1) * COUT];

        // Keep the weight stream ahead (~8KB): global_prefetch_b8, speculative.
        __builtin_prefetch(Wp + (size_t)(k0 + 32) * COUT, 0, 1);

        // D = A*B + C  (8 args: neg_a, A, neg_b, B, c_mod, C, reuse_a, reuse_b)
        acc = __builtin_amdgcn_wmma_f32_16x16x4_f32(false, a, false, b,
                                                    (short)0, acc, false, false);
    }

    // ---- Store: real batches are M=0..7 -> VGPRs 0..7, lanes 0..15 (ISA C/D layout) ----
    if (lane < 16) {
        float* __restrict__ op = out + (size_t)loc * COUT + n0 + lane;
        #pragma unroll
        for (int i = 0; i < 8; ++i)
            op[(size_t)i * H_OUT * W_OUT * COUT] = acc[i];
    }
}

extern "C" void kernel_launch(void* const* d_in, const int* in_sizes, int n_in,
                              void* d_out, int out_size, void* d_ws, size_t ws_size,
                              hipStream_t stream)
{
    (void)in_sizes; (void)n_in; (void)d_ws; (void)ws_size; (void)out_size;
    const float* x  = (const float*)d_in[0];   // (8,32,32,64) fp32
    const float* wp = (const float*)d_in[1];   // (30,30,576,64) fp32
    float* out      = (float*)d_out;           // (8,30,30,64) fp32

    dim3 grid(H_OUT * W_OUT);  // 900 locations
    dim3 block(128);           // 4 wave32 -> 4 N-tiles of 16
    lc_wmma_kernel<<<grid, block, 0, stream>>>(x, wp, out);
}